// CriterionMCV_50414326120601
// MI455X (gfx1250) — compile-verified
//
#include <hip/hip_runtime.h>
#include <hip/hip_bf16.h>
#include <math.h>

#define N_ROWS 4096
#define A_DIM  512
#define C_CLS  100

typedef __attribute__((ext_vector_type(2))) float v2f;
typedef __attribute__((ext_vector_type(8))) float v8f;

// ---------------- kernel 0: zero class counts ----------------
__global__ void k_zero(int* __restrict__ counts) {
    int i = threadIdx.x;
    if (i < C_CLS) counts[i] = 0;
}

// ---------------- kernel 1: counts (int atomics, exact) + per-row NLL ----
__global__ void k_count_loss(const float* __restrict__ y_s,
                             const int* __restrict__ labels,
                             int* __restrict__ counts,
                             float* __restrict__ rowloss) {
    int n = blockIdx.x * blockDim.x + threadIdx.x;
    if (n >= N_ROWS) return;
    int lab = labels[n];
    atomicAdd(&counts[lab], 1);
    const float* row = y_s + (size_t)n * C_CLS;
    float m = row[0];
    #pragma unroll 4
    for (int c = 1; c < C_CLS; ++c) m = fmaxf(m, row[c]);
    float s = 0.f;
    #pragma unroll 4
    for (int c = 0; c < C_CLS; ++c) s += __expf(row[c] - m);
    rowloss[n] = -(row[lab] - m - __logf(s));
}

// ---------------- kernel 2: deterministic grouping of rows by class ------
__global__ void k_group(const int* __restrict__ labels,
                        const int* __restrict__ counts,
                        int* __restrict__ offsets,
                        int* __restrict__ order) {
    if (threadIdx.x == 0) {
        int acc = 0;
        for (int c = 0; c <= C_CLS; ++c) {
            offsets[c] = acc;
            if (c < C_CLS) acc += counts[c];
        }
    }
    __syncthreads();
    int c = threadIdx.x;
    if (c < C_CLS) {
        int pos = offsets[c];
        for (int n = 0; n < N_ROWS; ++n)
            if (labels[n] == c) order[pos++] = n;
    }
}

// ---------------- kernel 3: per-class mean + fused new_ave ---------------
__global__ void k_ave(const float* __restrict__ features,
                      const float* __restrict__ Ave_old,
                      const float* __restrict__ Amount,
                      const int* __restrict__ counts,
                      const int* __restrict__ offsets,
                      const int* __restrict__ order,
                      float* __restrict__ ave,
                      float* __restrict__ out_ave) {
    int c = blockIdx.y;
    int a = blockIdx.x * blockDim.x + threadIdx.x;
    if (a >= A_DIM) return;
    int beg = offsets[c], end = offsets[c + 1];
    float s = 0.f;
    for (int r = beg; r < end; ++r)
        s += features[(size_t)order[r] * A_DIM + a];
    int   cnt  = counts[c];
    float cntf = (float)cnt;
    float safe = (cnt == 0) ? 1.f : cntf;
    float av   = s / safe;
    ave[(size_t)c * A_DIM + a] = av;
    float denom = cntf + Amount[c];
    float w = (denom > 0.f) ? (cntf / denom) : 0.f;
    out_ave[(size_t)c * A_DIM + a] =
        Ave_old[(size_t)c * A_DIM + a] * (1.f - w) + av * w;
}

// ---------------- kernel 4: per-class scatter via f32 WMMA + fused cov update
// One wave32 computes a 32x32 macro-tile of G_c = X_c^T X_c as a 2x2 grid of
// 16x16 V_WMMA_F32_16X16X4_F32 tiles (4 accumulators, 4 WMMAs per K=4 step,
// A/B fragments reused across the 2x2 -> 8 loads feed 4096 MACs).
// A-frag (16x4 f32, ISA 7.12.2): lanes 0-15 K=0,1 ; lanes 16-31 K=2,3 ; M = lane&15.
// B-frag (4x16 f32): v0 = K=0 (lanes 0-15) / K=1 (lanes 16-31); v1 = K=2/K=3.
// C/D (16x16 f32, 8 VGPRs): VGPR v, lane l -> M = v + 8*(l>=16), N = l&15.
__global__ __launch_bounds__(32) void k_cov_wmma(
    const float* __restrict__ features,
    const float* __restrict__ cov_old,
    const float* __restrict__ Ave_old,
    const float* __restrict__ Amount,
    const int*   __restrict__ counts,
    const int*   __restrict__ offsets,
    const int*   __restrict__ order,
    const float* __restrict__ ave,
    float*       __restrict__ out_cov) {
    int ti = blockIdx.x, tj = blockIdx.y, c = blockIdx.z;
    int lane = threadIdx.x;        // 0..31, wave32
    int l15  = lane & 15;
    int hi   = lane >> 4;          // 0 or 1
    int ibase = ti * 32, jbase = tj * 32;
    int beg = offsets[c], end = offsets[c + 1];

    const float* aveC = ave + (size_t)c * A_DIM;
    float ave_i0 = aveC[ibase +      l15];
    float ave_i1 = aveC[ibase + 16 + l15];
    float ave_j0 = aveC[jbase +      l15];
    float ave_j1 = aveC[jbase + 16 + l15];

    v8f acc00 = {}, acc01 = {}, acc10 = {}, acc11 = {};

    for (int r = beg; r < end; r += 4) {
        // order[r..r+3] are wave-uniform -> scalar loads; clamp OOB to r
        // (loaded garbage rows are zeroed by the validity selects below).
        int o0 = order[r];
        int o1 = (r + 1 < end) ? order[r + 1] : o0;
        int o2 = (r + 2 < end) ? order[r + 2] : o0;
        int o3 = (r + 3 < end) ? order[r + 3] : o0;

        // per-half-wave row selection (A consumes K-pairs, B consumes K rows)
        int  oA0 = hi ? o2 : o0;
        int  oA1 = hi ? o3 : o1;
        int  oB0 = hi ? o1 : o0;
        int  oB1 = hi ? o3 : o2;
        bool vA0 = (r + 2 * hi)     < end;
        bool vA1 = (r + 2 * hi + 1) < end;
        bool vB0 = (r + hi)         < end;
        bool vB1 = (r + hi + 2)     < end;

        const float* rA0 = features + (size_t)oA0 * A_DIM;
        const float* rA1 = features + (size_t)oA1 * A_DIM;
        const float* rB0 = features + (size_t)oB0 * A_DIM;
        const float* rB1 = features + (size_t)oB1 * A_DIM;

        v2f af0, af1, bf0, bf1;
        af0.x = vA0 ? rA0[ibase +      l15] - ave_i0 : 0.f;
        af0.y = vA1 ? rA1[ibase +      l15] - ave_i0 : 0.f;
        af1.x = vA0 ? rA0[ibase + 16 + l15] - ave_i1 : 0.f;
        af1.y = vA1 ? rA1[ibase + 16 + l15] - ave_i1 : 0.f;
        bf0.x = vB0 ? rB0[jbase +      l15] - ave_j0 : 0.f;
        bf0.y = vB1 ? rB1[jbase +      l15] - ave_j0 : 0.f;
        bf1.x = vB0 ? rB0[jbase + 16 + l15] - ave_j1 : 0.f;
        bf1.y = vB1 ? rB1[jbase + 16 + l15] - ave_j1 : 0.f;

        acc00 = __builtin_amdgcn_wmma_f32_16x16x4_f32(false, af0, false, bf0, (short)0, acc00, false, false);
        acc01 = __builtin_amdgcn_wmma_f32_16x16x4_f32(false, af0, false, bf1, (short)0, acc01, false, false);
        acc10 = __builtin_amdgcn_wmma_f32_16x16x4_f32(false, af1, false, bf0, (short)0, acc10, false, false);
        acc11 = __builtin_amdgcn_wmma_f32_16x16x4_f32(false, af1, false, bf1, (short)0, acc11, false, false);
    }

    // fused EMA epilogue: new_cov = cov_old*(1-w) + (G/counts_safe)*w + w(1-w)*diff_i*diff_j
    int   cnt   = counts[c];
    float cntf  = (float)cnt;
    float safe  = (cnt == 0) ? 1.f : cntf;
    float denom = cntf + Amount[c];
    float w     = (denom > 0.f) ? (cntf / denom) : 0.f;
    float scale = w / safe;
    float omw   = 1.f - w;
    float ww    = w * omw;

    const float* AveC = Ave_old + (size_t)c * A_DIM;
    float diff_j0 = AveC[jbase +      l15] - ave_j0;
    float diff_j1 = AveC[jbase + 16 + l15] - ave_j1;
    size_t covbase = (size_t)c * A_DIM * A_DIM;

    #pragma unroll
    for (int p = 0; p < 2; ++p) {
        const v8f& aL = p ? acc10 : acc00;
        const v8f& aR = p ? acc11 : acc01;
        #pragma unroll
        for (int v = 0; v < 8; ++v) {
            int row = ibase + 16 * p + v + 8 * hi;
            float diff_i = AveC[row] - aveC[row];
            size_t rowoff = covbase + (size_t)row * A_DIM;
            int col0 = jbase +      l15;
            int col1 = jbase + 16 + l15;
            out_cov[rowoff + col0] = cov_old[rowoff + col0] * omw + aL[v] * scale + ww * diff_i * diff_j0;
            out_cov[rowoff + col1] = cov_old[rowoff + col1] * omw + aR[v] * scale + ww * diff_i * diff_j1;
        }
    }
}

// ---------------- kernel 5: fixed-order loss reduction + new_amount ------
__global__ void k_final(const float* __restrict__ rowloss,
                        const float* __restrict__ Amount,
                        const int*   __restrict__ counts,
                        float* __restrict__ out_loss,
                        float* __restrict__ out_amount) {
    __shared__ float sm[256];
    int t = threadIdx.x;
    float s = 0.f;
    for (int n = t; n < N_ROWS; n += 256) s += rowloss[n];   // fixed strided order
    sm[t] = s;
    __syncthreads();
    for (int step = 128; step > 0; step >>= 1) {             // fixed tree order
        if (t < step) sm[t] += sm[t + step];
        __syncthreads();
    }
    if (t == 0) out_loss[0] = sm[0] / (float)N_ROWS;
    if (t < C_CLS) out_amount[t] = Amount[t] + (float)counts[t];
}

// ---------------- launcher ----------------
extern "C" void kernel_launch(void* const* d_in, const int* in_sizes, int n_in,
                              void* d_out, int out_size, void* d_ws, size_t ws_size,
                              hipStream_t stream) {
    const float* features = (const float*)d_in[0];   // [N, A]
    const float* y_s      = (const float*)d_in[1];   // [N, C]
    const float* CoVar    = (const float*)d_in[2];   // [C, A, A]
    const float* Ave      = (const float*)d_in[3];   // [C, A]
    const float* Amount   = (const float*)d_in[4];   // [C]
    const int*   labels   = (const int*)d_in[5];     // [N]

    // workspace carve-up (~240 KB)
    char* w = (char*)d_ws;
    int*   counts  = (int*)w;     w += 512;
    int*   offsets = (int*)w;     w += 512;
    int*   order   = (int*)w;     w += (size_t)N_ROWS * 4;
    float* rowloss = (float*)w;   w += (size_t)N_ROWS * 4;
    float* ave     = (float*)w;   w += (size_t)C_CLS * A_DIM * 4;

    // output carve-up: (loss, new_cov, new_ave, new_amount) flat
    float* out        = (float*)d_out;
    float* out_loss   = out;
    float* out_cov    = out + 1;
    float* out_ave    = out + 1 + (size_t)C_CLS * A_DIM * A_DIM;
    float* out_amount = out_ave + (size_t)C_CLS * A_DIM;

    k_zero<<<1, 128, 0, stream>>>(counts);
    k_count_loss<<<(N_ROWS + 255) / 256, 256, 0, stream>>>(y_s, labels, counts, rowloss);
    k_group<<<1, 128, 0, stream>>>(labels, counts, offsets, order);
    k_ave<<<dim3(A_DIM / 64, C_CLS), 64, 0, stream>>>(
        features, Ave, Amount, counts, offsets, order, ave, out_ave);
    k_cov_wmma<<<dim3(A_DIM / 32, A_DIM / 32, C_CLS), 32, 0, stream>>>(
        features, CoVar, Ave, Amount, counts, offsets, order, ave, out_cov);
    k_final<<<1, 256, 0, stream>>>(rowloss, Amount, counts, out_loss, out_amount);
}